// h_diag_Land_63453846831510
// MI455X (gfx1250) — compile-verified
//
#include <hip/hip_runtime.h>
#include <hip/hip_bf16.h>

typedef float v2f __attribute__((ext_vector_type(2)));
typedef float v8f __attribute__((ext_vector_type(8)));

#define DDIM   128
#define BT     16      // b-tile rows per block
#define NSUP   128     // n's per super-tile (8 waves x 16)
#define XR_STR 132     // padded LDS row stride (floats) -> conflict-free transposed reads
#define WS_STR 20      // padded W-tile stride (floats)
#define WT_SZ  (16 * WS_STR)
#define INV_2G2 12.5f  // 1 / (2 * 0.2^2)

__launch_bounds__(256)
__global__ void rbf_wsq_kernel(const float* __restrict__ x,
                               const float* __restrict__ r,
                               float* __restrict__ outbase,
                               int tiles_per_split,
                               unsigned long long bd) {
  __shared__ float Xs[BT * XR_STR];        // x tile, resident whole kernel
  __shared__ float Rs[NSUP * XR_STR];      // current 128-row r super-tile (~67.6 KB)
  __shared__ float Ws[8 * WT_SZ];          // 8 weight tiles (C->A layout bounce)
  __shared__ float rpart[NSUP * 2];        // ||r||^2 partials (2 halves per row)
  __shared__ float xpart[16 * 17];
  __shared__ float xsq[16];
  __shared__ float s0_lds[16];

  const int tid  = threadIdx.x;
  const int wave = tid >> 5;
  const int lane = tid & 31;
  const int lo   = lane & 15;
  const int hi   = lane >> 4;
  const int b0   = blockIdx.x * BT;
  const int dsl  = wave * 16;              // this wave's 16-wide d slice
  float* outp = outbase + (unsigned long long)blockIdx.y * bd;

  // ---- load X tile into LDS, compute ||x||^2 ----
  {
    const int row = tid >> 4;
    const int c8  = (tid & 15) * 8;
    const float4* gx = (const float4*)(x + (b0 + row) * DDIM + c8);
    float4 v0 = gx[0], v1 = gx[1];
    float* dst = Xs + row * XR_STR + c8;
    ((float4*)dst)[0] = v0;
    ((float4*)(dst + 4))[0] = v1;
    xpart[row * 17 + (tid & 15)] =
        v0.x*v0.x + v0.y*v0.y + v0.z*v0.z + v0.w*v0.w +
        v1.x*v1.x + v1.y*v1.y + v1.z*v1.z + v1.w*v1.w;
  }
  if (tid < 16) s0_lds[tid] = 0.0f;
  __syncthreads();
  if (tid < 16) {
    float s = 0.0f;
#pragma unroll
    for (int j = 0; j < 16; ++j) s += xpart[tid * 17 + j];
    xsq[tid] = s;
  }
  __syncthreads();

  // four independent WMMA accumulation chains in the consumer stage
  v8f s1e = {}, s1o = {};             // Sum_n w * r    (16b x 16d tile, C layout)
  v8f s2e = {}, s2o = {};             // Sum_n w * r^2
  float s0p[8] = {0,0,0,0,0,0,0,0};   // per-lane partial of Sum_n w

  const int t0 = blockIdx.y * tiles_per_split;
  for (int t = t0; t < t0 + tiles_per_split; ++t) {
    // ---- phase 1: cooperative load of 128x128 R super-tile + ||r||^2 partials ----
    {
      const int row  = tid >> 1;          // 2 threads per row
      const int half = tid & 1;
      const float4* gr = (const float4*)(r + ((t * NSUP + row) * DDIM) + half * 64);
      float* dst = Rs + row * XR_STR + half * 64;
      float p = 0.0f;
#pragma unroll
      for (int i = 0; i < 16; ++i) {
        float4 v = gr[i];
        ((float4*)(dst + i * 4))[0] = v;
        p += v.x*v.x + v.y*v.y + v.z*v.z + v.w*v.w;
      }
      rpart[row * 2 + half] = p;
    }
    __syncthreads();

    // ---- phase 2: ALL waves produce in parallel: wave w -> W tile for n-subtile w ----
    {
      const int nrow = wave * 16 + lo;    // column n of G this lane covers (within subtile)
      float rq = rpart[nrow * 2] + rpart[nrow * 2 + 1];

      v8f g0 = {}, g1 = {};               // two independent chains over even/odd K-chunks
#pragma unroll
      for (int k0 = 0; k0 < DDIM; k0 += 8) {
        // A (16x4 f32): lanes0-15 row=lo K=k0,k0+1 ; lanes16-31 row=lo K=k0+2,k0+3
        v2f a0 = *(const v2f*)(Xs + lo * XR_STR + k0 + 2 * hi);
        v2f b0 = *(const v2f*)(Rs + nrow * XR_STR + k0 + 2 * hi);
        v2f a1 = *(const v2f*)(Xs + lo * XR_STR + k0 + 4 + 2 * hi);
        v2f b1 = *(const v2f*)(Rs + nrow * XR_STR + k0 + 4 + 2 * hi);
        g0 = __builtin_amdgcn_wmma_f32_16x16x4_f32(false, a0, false, b0,
                                                   (short)0, g0, false, false);
        g1 = __builtin_amdgcn_wmma_f32_16x16x4_f32(false, a1, false, b1,
                                                   (short)0, g1, false, false);
      }
      v8f g = g0 + g1;
#pragma unroll
      for (int v = 0; v < 8; ++v) {
        const int brow = v + 8 * hi;      // C layout: M = v + 8*hi, N = lo
        float dist = xsq[brow] + rq - 2.0f * g[v];
        float w = __expf(-INV_2G2 * dist);
        s0p[v] += w;
        Ws[wave * WT_SZ + brow * WS_STR + lo] = w;
      }
    }
    __syncthreads();

    // ---- phase 3: all waves consume all 8 W tiles for their own d slice ----
    for (int wt = 0; wt < 8; wt += 2) {
#pragma unroll
      for (int kc = 0; kc < 4; ++kc) {
        const int k0 = kc * 4;
        // even wt -> chains s1e/s2e
        v2f ae = *(const v2f*)(Ws + wt * WT_SZ + lo * WS_STR + k0 + 2 * hi);
        const int re = wt * 16 + k0 + 2 * hi;
        float e0 = Rs[re * XR_STR + dsl + lo];
        float e1 = Rs[(re + 1) * XR_STR + dsl + lo];
        // odd wt -> chains s1o/s2o
        v2f ao = *(const v2f*)(Ws + (wt + 1) * WT_SZ + lo * WS_STR + k0 + 2 * hi);
        const int ro = (wt + 1) * 16 + k0 + 2 * hi;
        float o0 = Rs[ro * XR_STR + dsl + lo];
        float o1 = Rs[(ro + 1) * XR_STR + dsl + lo];

        v2f ble = {e0, e1};
        s1e = __builtin_amdgcn_wmma_f32_16x16x4_f32(false, ae, false, ble,
                                                    (short)0, s1e, false, false);
        v2f blo = {o0, o1};
        s1o = __builtin_amdgcn_wmma_f32_16x16x4_f32(false, ao, false, blo,
                                                    (short)0, s1o, false, false);
        v2f bqe = {e0 * e0, e1 * e1};
        s2e = __builtin_amdgcn_wmma_f32_16x16x4_f32(false, ae, false, bqe,
                                                    (short)0, s2e, false, false);
        v2f bqo = {o0 * o0, o1 * o1};
        s2o = __builtin_amdgcn_wmma_f32_16x16x4_f32(false, ao, false, bqo,
                                                    (short)0, s2o, false, false);
      }
    }
    __syncthreads();
  }

  v8f s1 = s1e + s1o;
  v8f s2 = s2e + s2o;

  // ---- reduce S0 across waves (each wave covered only its own 16-n subtile) ----
#pragma unroll
  for (int v = 0; v < 8; ++v)
    atomicAdd(&s0_lds[v + 8 * hi], s0p[v]);
  __syncthreads();

  // ---- partial combine: out_partial = S2 - 2*x*S1 + x^2*S0 (exact, additive in n) ----
#pragma unroll
  for (int v = 0; v < 8; ++v) {
    const int brow = v + 8 * hi;
    const int d    = dsl + lo;
    float xv = Xs[brow * XR_STR + d];
    float o  = s2[v] - 2.0f * xv * s1[v] + xv * xv * s0_lds[brow];
    outp[(unsigned long long)(b0 + brow) * DDIM + d] = o;
  }
}

// deterministic fixed-order reduction of the per-split partial planes
__global__ void combine_kernel(float* __restrict__ out,
                               const float* __restrict__ ws,
                               int nsplit, unsigned long long bd) {
  unsigned long long i = (unsigned long long)blockIdx.x * blockDim.x + threadIdx.x;
  if (i < bd) {
    float s = 0.0f;
    for (int k = 0; k < nsplit; ++k) s += ws[(unsigned long long)k * bd + i];
    out[i] = s;
  }
}

extern "C" void kernel_launch(void* const* d_in, const int* in_sizes, int n_in,
                              void* d_out, int out_size, void* d_ws, size_t ws_size,
                              hipStream_t stream) {
  const float* x = (const float*)d_in[0];
  const float* r = (const float*)d_in[1];
  float* out = (float*)d_out;

  const int B = in_sizes[0] / DDIM;            // 1024
  const int N = in_sizes[1] / DDIM;            // 2048
  const int nsuper = N / NSUP;                 // 16 super-tiles
  const unsigned long long bd = (unsigned long long)B * DDIM;

  int nsplit = 1;
  if ((nsuper % 4 == 0) && ws_size >= 4ull * bd * sizeof(float))      nsplit = 4;
  else if ((nsuper % 2 == 0) && ws_size >= 2ull * bd * sizeof(float)) nsplit = 2;

  dim3 block(256);
  if (nsplit == 1) {
    rbf_wsq_kernel<<<dim3(B / BT, 1), block, 0, stream>>>(x, r, out, nsuper, bd);
  } else {
    rbf_wsq_kernel<<<dim3(B / BT, nsplit), block, 0, stream>>>(
        x, r, (float*)d_ws, nsuper / nsplit, bd);
    unsigned long long nblk = (bd + 255) / 256;
    combine_kernel<<<dim3((unsigned)nblk), block, 0, stream>>>(out, (const float*)d_ws,
                                                               nsplit, bd);
  }
}